// OT_Loss_Geo_28759101014106
// MI455X (gfx1250) — compile-verified
//
#include <hip/hip_runtime.h>

// ---------------------------------------------------------------------------
// OT_Loss_Geo (debiased unbalanced Sinkhorn, geomloss-style) for MI455X.
// One persistent workgroup per image; all state LDS-resident; cost Gram tiles
// computed with V_WMMA_F32_16X16X4_F32 (K=2 coords padded to 4); per-tile
// block-max logsumexp: one rescale + 16 independent v_exp_f32 per 2 WMMAs
// so the TRANS pipe co-issues against the matrix pipe.
// ---------------------------------------------------------------------------

typedef float v2f __attribute__((ext_vector_type(2)));
typedef float v8f __attribute__((ext_vector_type(8)));

#define NG 1024          // grid points (32x32)
#define NP 512           // annotated points
#define NBATCH 8
#define RHO_F 0.25f      // reach^2
#define NEPS 11

// geomloss epsilon schedule: [D^2] + exp(arange(2logD, 2logBlur, 2logScaling)) + [blur^2]
__device__ __constant__ float c_eps[NEPS] = {
    8.f, 8.f, 2.f, 0.5f, 0.125f, 0.03125f, 0.0078125f,
    0.001953125f, 0.00048828125f, 0.0001220703125f, 1e-4f};

// hh[j] = (alog[j]?) + pot[j]/eps - 0.5*|c1_j|^2/eps   (reduction-side vector)
__device__ __noinline__ void build_h(float* hh,
                                     const float* alogArr, const float* potArr,
                                     const float* nrm, int n, float inv_eps)
{
    for (int j = threadIdx.x; j < n; j += blockDim.x) {
        float h = alogArr ? alogArr[j] : 0.f;
        if (potArr) h = fmaf(potArr[j], inv_eps, h);
        hh[j] = fmaf(-0.5f * nrm[j], inv_eps, h);
    }
}

// out[i] = coefOld*oldArr[i] + coefNew * ( -eps * logsumexp_j( hh[j] - 0.5|c2_i|^2/eps + c1_j.c2_i/eps ) )
// Rows of A = reduction index j (c1), cols of B = output index i (c2).
// D-tile: lane l -> N = l&15 (output i), VGPR m -> M = m + 8*(l>>4) (reduced j).
__device__ __noinline__ void softmin_pass(
    const float* c1x, const float* c1y, const float* hh, int nJ,
    const float* c2x, const float* c2y, const float* n2, int nI,
    float eps, float inv_eps,
    float coefOld, const float* oldArr, float coefNew, float* out)
{
    const int  lane = threadIdx.x & 31;
    const int  wave = threadIdx.x >> 5;
    const int  nwav = blockDim.x >> 5;
    const bool lo   = lane < 16;
    const int  nidx = lane & 15;
    const int  moff = (lane >> 4) << 3;      // 0 or 8

    for (int it = wave * 16; it < nI; it += nwav * 16) {
        const int i = it + nidx;
        // B fragment (4x16 f32): lanes 0-15 carry K=0,1 of column N=lane; lanes 16-31 K=2,3 = 0 pad
        v2f bfrag;
        bfrag.x = lo ? c2x[i] : 0.f;
        bfrag.y = lo ? c2y[i] : 0.f;
        const float cN = -0.5f * n2[i] * inv_eps;   // per-output-row constant, added post-LSE
        float mx = -1e30f, sm = 0.f;

        for (int jt = 0; jt < nJ; jt += 32) {       // 2 WMMA tiles per iteration
            // A fragments (16x4 f32): lanes 0-15 carry K=0,1 of row M=lane; lanes 16-31 zero pad
            v2f a0, a1;
            a0.x = lo ? c1x[jt + nidx]      : 0.f;
            a0.y = lo ? c1y[jt + nidx]      : 0.f;
            a1.x = lo ? c1x[jt + 16 + nidx] : 0.f;
            a1.y = lo ? c1y[jt + 16 + nidx] : 0.f;
            v8f z = {0.f, 0.f, 0.f, 0.f, 0.f, 0.f, 0.f, 0.f};
            v8f acc0 = __builtin_amdgcn_wmma_f32_16x16x4_f32(
                false, a0, false, bfrag, (short)0, z, false, false);
            v8f acc1 = __builtin_amdgcn_wmma_f32_16x16x4_f32(
                false, a1, false, bfrag, (short)0, z, false, false);

            const float* h0 = hh + jt + moff;
            const float* h1 = hh + jt + 16 + moff;
            float v[16];
#pragma unroll
            for (int m = 0; m < 8; ++m) {
                v[m]     = fmaf(acc0[m], inv_eps, h0[m]);   // h_j - C_ij/eps (minus cN)
                v[m + 8] = fmaf(acc1[m], inv_eps, h1[m]);
            }
            // depth-4 tree max over the 16 fresh values
            float t0 = fmaxf(fmaxf(v[0],  v[1]),  fmaxf(v[2],  v[3]));
            float t1 = fmaxf(fmaxf(v[4],  v[5]),  fmaxf(v[6],  v[7]));
            float t2 = fmaxf(fmaxf(v[8],  v[9]),  fmaxf(v[10], v[11]));
            float t3 = fmaxf(fmaxf(v[12], v[13]), fmaxf(v[14], v[15]));
            float nm = fmaxf(mx, fmaxf(fmaxf(t0, t1), fmaxf(t2, t3)));
            // 16 independent exps (TRANS pipe), tree-summed
            float e0 = __expf(v[0]  - nm) + __expf(v[1]  - nm);
            float e1 = __expf(v[2]  - nm) + __expf(v[3]  - nm);
            float e2 = __expf(v[4]  - nm) + __expf(v[5]  - nm);
            float e3 = __expf(v[6]  - nm) + __expf(v[7]  - nm);
            float e4 = __expf(v[8]  - nm) + __expf(v[9]  - nm);
            float e5 = __expf(v[10] - nm) + __expf(v[11] - nm);
            float e6 = __expf(v[12] - nm) + __expf(v[13] - nm);
            float e7 = __expf(v[14] - nm) + __expf(v[15] - nm);
            float s  = ((e0 + e1) + (e2 + e3)) + ((e4 + e5) + (e6 + e7));
            // single rescale of the running sum
            sm = fmaf(sm, __expf(mx - nm), s);
            mx = nm;
        }
        // combine the two half-wave partials covering M=0..7 and M=8..15 of each tile
        float mo = __shfl_xor(mx, 16, 32);
        float so = __shfl_xor(sm, 16, 32);
        float nm = fmaxf(mx, mo);
        float s  = sm * __expf(mx - nm) + so * __expf(mo - nm);
        float r  = -eps * (nm + cN + __logf(s));
        if (lo) {
            float o = (coefOld != 0.f) ? coefOld * oldArr[i] : 0.f;
            out[i] = o + coefNew * r;
        }
    }
}

__device__ float block_sum(float v, float* red)
{
    __syncthreads();                                  // protect red reuse
    for (int o = 16; o > 0; o >>= 1) v += __shfl_xor(v, o, 32);
    const int lane = threadIdx.x & 31, wave = threadIdx.x >> 5, nw = blockDim.x >> 5;
    if (lane == 0) red[wave] = v;
    __syncthreads();
    float t = (lane < nw) ? red[lane] : 0.f;
    for (int o = 16; o > 0; o >>= 1) t += __shfl_xor(t, o, 32);
    return t;
}

__global__ void __launch_bounds__(256)
ot_sinkhorn(const float* __restrict__ unnormed, const float* __restrict__ pts,
            float* __restrict__ partial)
{
    __shared__ float gx[NG], gy[NG], nx[NG], alog[NG];
    __shared__ float f_aa[NG], f_ba[NG], ft[NG], hh[NG];
    __shared__ float yx[NP], yy[NP], ny[NP], g_bb[NP], g_ab[NP];
    __shared__ float red[32];

    const int b   = blockIdx.x;
    const int tid = threadIdx.x;
    const int nt  = blockDim.x;

    // grid: c_k = (8k+4)/256*2-1 ; grid[i] = (c[i&31], c[i>>5])  (dstack((Y,X)))
    for (int i = tid; i < NG; i += nt) {
        float c0 = (float)(i & 31) * 0.0625f + (0.03125f - 1.f);
        float c1 = (float)(i >> 5) * 0.0625f + (0.03125f - 1.f);
        gx[i] = c0; gy[i] = c1; nx[i] = c0 * c0 + c1 * c1;
    }
    // points: normalized twice (faithful to the source)
    const float* p = pts + (size_t)b * NP * 2;
    for (int m = tid; m < NP; m += nt) {
        float q0 = p[2 * m + 0] * (2.f / 256.f) - 1.f; q0 = q0 * (2.f / 256.f) - 1.f;
        float q1 = p[2 * m + 1] * (2.f / 256.f) - 1.f; q1 = q1 * (2.f / 256.f) - 1.f;
        yx[m] = q0; yy[m] = q1; ny[m] = q0 * q0 + q1 * q1;
    }
    // a_log = log((u+1)/sum(u+1))
    const float* u = unnormed + (size_t)b * NG;
    float ps = 0.f;
    for (int i = tid; i < NG; i += nt) ps += u[i] + 1.f;
    float total  = block_sum(ps, red);
    float logtot = __logf(total);
    for (int i = tid; i < NG; i += nt) alog[i] = __logf(u[i] + 1.f) - logtot;
    __syncthreads();

    // ---- init at eps0 ----
    {
        float eps = c_eps[0], ie = 1.f / eps, d = 1.f / (1.f + eps / RHO_F);
        build_h(hh, alog, nullptr, nx, NG, ie); __syncthreads();
        softmin_pass(gx, gy, hh, NG, gx, gy, nx, NG, eps, ie, 0.f, nullptr, d, f_aa);
        softmin_pass(gx, gy, hh, NG, yx, yy, ny, NP, eps, ie, 0.f, nullptr, d, g_ab);
        __syncthreads();
        build_h(hh, nullptr, nullptr, ny, NP, ie); __syncthreads();
        softmin_pass(yx, yy, hh, NP, yx, yy, ny, NP, eps, ie, 0.f, nullptr, d, g_bb);
        softmin_pass(yx, yy, hh, NP, gx, gy, nx, NG, eps, ie, 0.f, nullptr, d, f_ba);
        __syncthreads();
    }

    // ---- epsilon-scaling loop, symmetric averaged updates ----
    for (int k = 0; k < NEPS; ++k) {
        float eps = c_eps[k], ie = 1.f / eps, d = 1.f / (1.f + eps / RHO_F);
        // ft_ba = d * softmin(eps, Cxy, b_log + g_ab/eps)
        build_h(hh, nullptr, g_ab, ny, NP, ie); __syncthreads();
        softmin_pass(yx, yy, hh, NP, gx, gy, nx, NG, eps, ie, 0.f, nullptr, d, ft); __syncthreads();
        // gt_ab = d * softmin(eps, Cyx, a_log + f_ba/eps)  (old f_ba)
        build_h(hh, alog, f_ba, nx, NG, ie); __syncthreads();
        softmin_pass(gx, gy, hh, NG, yx, yy, ny, NP, eps, ie, 0.5f, g_ab, 0.5f * d, g_ab); __syncthreads();
        for (int i = tid; i < NG; i += nt) f_ba[i] = 0.5f * (f_ba[i] + ft[i]);
        __syncthreads();
        // f_aa = 0.5*(f_aa + d*softmin(eps, Cxx, a_log + f_aa/eps))
        build_h(hh, alog, f_aa, nx, NG, ie); __syncthreads();
        softmin_pass(gx, gy, hh, NG, gx, gy, nx, NG, eps, ie, 0.5f, f_aa, 0.5f * d, f_aa); __syncthreads();
        // g_bb = 0.5*(g_bb + d*softmin(eps, Cyy, b_log + g_bb/eps))
        build_h(hh, nullptr, g_bb, ny, NP, ie); __syncthreads();
        softmin_pass(yx, yy, hh, NP, yx, yy, ny, NP, eps, ie, 0.5f, g_bb, 0.5f * d, g_bb); __syncthreads();
    }

    // ---- final extrapolation at eps = blur^p ----
    {
        float eps = c_eps[NEPS - 1], ie = 1.f / eps, d = 1.f / (1.f + eps / RHO_F);
        build_h(hh, nullptr, g_ab, ny, NP, ie); __syncthreads();
        softmin_pass(yx, yy, hh, NP, gx, gy, nx, NG, eps, ie, 0.f, nullptr, d, ft); __syncthreads();   // f_ba_new
        build_h(hh, alog, f_ba, nx, NG, ie); __syncthreads();                                          // old f_ba
        softmin_pass(gx, gy, hh, NG, yx, yy, ny, NP, eps, ie, 0.f, nullptr, d, g_ab); __syncthreads(); // g_ab_new
        build_h(hh, alog, f_aa, nx, NG, ie); __syncthreads();
        softmin_pass(gx, gy, hh, NG, gx, gy, nx, NG, eps, ie, 0.f, nullptr, d, f_aa); __syncthreads();
        build_h(hh, nullptr, g_bb, ny, NP, ie); __syncthreads();
        softmin_pass(yx, yy, hh, NP, yx, yy, ny, NP, eps, ie, 0.f, nullptr, d, g_bb); __syncthreads();

        // loss = w * ( sum_i a_i (e^{-f_aa/rho} - e^{-f_ba/rho}) + sum_j (e^{-g_bb/rho} - e^{-g_ab/rho}) )
        const float inv_rho = 1.f / RHO_F;
        float acc = 0.f;
        for (int i = tid; i < NG; i += nt)
            acc += __expf(alog[i]) * (__expf(-f_aa[i] * inv_rho) - __expf(-ft[i] * inv_rho));
        for (int j = tid; j < NP; j += nt)
            acc += __expf(-g_bb[j] * inv_rho) - __expf(-g_ab[j] * inv_rho);
        float tot = block_sum(acc, red);
        if (tid == 0) partial[b] = (RHO_F + eps * 0.5f) * tot;
    }
}

__global__ void ot_finalize(const float* __restrict__ partial, float* __restrict__ out)
{
    if (threadIdx.x == 0 && blockIdx.x == 0) {
        float s = 0.f;
        for (int i = 0; i < NBATCH; ++i) s += partial[i];   // fixed order: deterministic
        out[0] = s;
    }
}

extern "C" void kernel_launch(void* const* d_in, const int* in_sizes, int n_in,
                              void* d_out, int out_size, void* d_ws, size_t ws_size,
                              hipStream_t stream)
{
    (void)in_sizes; (void)n_in; (void)out_size; (void)ws_size;
    // d_in[0] = normed_density (unused by reference), d_in[1] = unnormed_density, d_in[2] = points
    const float* unnormed = (const float*)d_in[1];
    const float* pts      = (const float*)d_in[2];
    float* partial        = (float*)d_ws;            // 8 floats of scratch

    ot_sinkhorn<<<NBATCH, 256, 0, stream>>>(unnormed, pts, partial);
    ot_finalize<<<1, 32, 0, stream>>>(partial, (float*)d_out);
}